// MultiBoxLoss_6141803233700
// MI455X (gfx1250) — compile-verified
//
#include <hip/hip_runtime.h>
#include <stdint.h>

#define B_ 128
#define P_ 8732
#define C_ 21
#define M_ 16
#define TPB 256
#define PCHUNKS ((P_ + TPB - 1) / TPB)   /* 35 */
#define NBLK3 (B_ * PCHUNKS)             /* 4480 = 70*64 */
#define NBLK3_PAD ((NBLK3 + 63) & ~63)

typedef __attribute__((ext_vector_type(2))) float v2f;
typedef __attribute__((ext_vector_type(8))) float v8f;

#if __has_builtin(__builtin_amdgcn_global_load_async_to_lds_b128)
#define HAVE_ASYNC_LDS 1
#else
#define HAVE_ASYNC_LDS 0
#endif

__device__ __forceinline__ float smooth_l1(float d) {
  float ad = fabsf(d);
  return (ad < 1.0f) ? 0.5f * d * d : ad - 0.5f;
}

// ---------------------------------------------------------------------------
// Kernel 1: per (b,p) jaccard vs 16 truths. Produces best_truth_overlap (bto),
// best_truth_idx (bti, first-occurrence argmax), and per-truth best prior via
// packed (iou_bits, ~p) atomicMax (ties -> lowest prior index, matching jnp.argmax).
// ---------------------------------------------------------------------------
__global__ void k_match(const float* __restrict__ priors,
                        const float* __restrict__ targets,
                        float* __restrict__ bto, int* __restrict__ bti,
                        unsigned long long* __restrict__ bpi) {
  __shared__ float t[M_ * 4];
  __shared__ float tarea[M_];
  __shared__ unsigned long long lbest[M_];
  const int b = blockIdx.y;
  const int tid = threadIdx.x;
  if (tid < M_ * 4) {
    int m = tid >> 2, c = tid & 3;
    t[tid] = targets[(size_t)(b * M_ + m) * 5 + c];
  }
  if (tid < M_) lbest[tid] = 0ull;
  __syncthreads();
  if (tid < M_)
    tarea[tid] = (t[tid * 4 + 2] - t[tid * 4 + 0]) * (t[tid * 4 + 3] - t[tid * 4 + 1]);
  __syncthreads();

  const int p = blockIdx.x * TPB + tid;
  if (p < P_) {
    float cx = priors[p * 4 + 0], cy = priors[p * 4 + 1];
    float w = priors[p * 4 + 2], h = priors[p * 4 + 3];
    float x0 = cx - 0.5f * w, y0 = cy - 0.5f * h;
    float x1 = cx + 0.5f * w, y1 = cy + 0.5f * h;
    float ap = w * h;
    float best = -1.0f; int bestm = 0;
    for (int m = 0; m < M_; ++m) {
      float lx = fmaxf(t[m * 4 + 0], x0);
      float ly = fmaxf(t[m * 4 + 1], y0);
      float rx = fminf(t[m * 4 + 2], x1);
      float ry = fminf(t[m * 4 + 3], y1);
      float iw = fmaxf(rx - lx, 0.0f);
      float ih = fmaxf(ry - ly, 0.0f);
      float inter = iw * ih;
      float iou = inter / (tarea[m] + ap - inter);
      if (iou > best) { best = iou; bestm = m; }  // strict > : first-occurrence argmax
      unsigned long long pk =
          ((unsigned long long)__float_as_uint(iou) << 32) |
          (unsigned long long)(unsigned int)(~(unsigned int)p);
      atomicMax(&lbest[m], pk);
    }
    bto[(size_t)b * P_ + p] = best;
    bti[(size_t)b * P_ + p] = bestm;
  }
  __syncthreads();
  if (tid < M_) atomicMax(&bpi[b * M_ + tid], lbest[tid]);
}

// ---------------------------------------------------------------------------
// Kernel 2: forced matches (sequential over m, last m wins on duplicate prior).
// ---------------------------------------------------------------------------
__global__ void k_force(const unsigned long long* __restrict__ bpi,
                        float* __restrict__ bto, int* __restrict__ bti) {
  int b = blockIdx.x * blockDim.x + threadIdx.x;
  if (b >= B_) return;
  for (int m = 0; m < M_; ++m) {
    unsigned long long pk = bpi[b * M_ + m];
    int p = (int)(~(unsigned int)(pk & 0xFFFFFFFFull));
    bto[(size_t)b * P_ + p] = 2.0f;
    bti[(size_t)b * P_ + p] = m;
  }
}

// ---------------------------------------------------------------------------
// Kernel 3: main streaming pass. Async-stage the conf chunk into LDS, compute
// logsumexp/CE, smooth-L1 on positives, per-block partial sums.
// val encoding: pos ? (-ce - 1) : ce   (so sign carries the pos flag).
// ---------------------------------------------------------------------------
__global__ void k_main(const float* __restrict__ loc,
                       const float* __restrict__ conf,
                       const float* __restrict__ priors,
                       const float* __restrict__ targets,
                       const float* __restrict__ bto, const int* __restrict__ bti,
                       float* __restrict__ val,
                       float* __restrict__ partial_l, float* __restrict__ partial_c,
                       int* __restrict__ num_pos) {
  __shared__ __align__(16) float sconf[TPB * C_];  // 21504 B
  __shared__ float starg[M_ * 5];
  __shared__ float redf[TPB];
  __shared__ float redf2[TPB];
  __shared__ int redi[TPB];

  const int b = blockIdx.y;
  const int tid = threadIdx.x;
  const int p0 = blockIdx.x * TPB;
  const int cnt_p = min(TPB, P_ - p0);
  const int n16 = (cnt_p * C_) / 4;  // 16B units (cnt_p is a multiple of 4)

  if (tid < M_ * 5) starg[tid] = targets[(size_t)b * M_ * 5 + tid];

  const char* gsrc = (const char*)(conf + ((size_t)b * P_ + (size_t)p0) * C_);
#if HAVE_ASYNC_LDS
  {
    typedef int vsi4 __attribute__((vector_size(16)));
    typedef __attribute__((address_space(1))) vsi4* gvp_t;
    typedef __attribute__((address_space(3))) vsi4* lvp_t;
    gvp_t gvp = (gvp_t)gsrc;
    lvp_t lvp = (lvp_t)sconf;
    for (int i = tid; i < n16; i += TPB)
      __builtin_amdgcn_global_load_async_to_lds_b128(gvp + i, lvp + i, 0, 0);
#if __has_builtin(__builtin_amdgcn_s_wait_asynccnt)
    __builtin_amdgcn_s_wait_asynccnt(0);
#else
    asm volatile("s_wait_asynccnt 0" ::: "memory");
#endif
  }
#else
  for (int i = tid; i < n16; i += TPB)
    ((float4*)sconf)[i] = ((const float4*)gsrc)[i];
#endif
  __syncthreads();

  const int p = p0 + tid;
  float lossl = 0.0f, cep = 0.0f; int pc = 0;
  if (p < P_) {
    const float* row = &sconf[tid * C_];
    float mx = row[0];
    for (int c = 1; c < C_; ++c) mx = fmaxf(mx, row[c]);
    float s = 0.0f;
    for (int c = 0; c < C_; ++c) s += expf(row[c] - mx);
    float lse = mx + logf(s);

    int ti = bti[(size_t)b * P_ + p];
    float ov = bto[(size_t)b * P_ + p];
    int lab = (int)starg[ti * 5 + 4];
    int ct = (ov < 0.5f) ? 0 : (lab + 1);
    float ce = lse - row[ct];
    bool pos = ct > 0;
    val[(size_t)b * P_ + p] = pos ? (-ce - 1.0f) : ce;

    if (pos) {
      pc = 1;
      cep = ce;  // positives always belong to the conf-loss mask
      float4 pr = ((const float4*)priors)[p];
      float m0 = starg[ti * 5 + 0], m1 = starg[ti * 5 + 1];
      float m2 = starg[ti * 5 + 2], m3 = starg[ti * 5 + 3];
      float gx = ((m0 + m2) * 0.5f - pr.x) / (0.1f * pr.z);
      float gy = ((m1 + m3) * 0.5f - pr.y) / (0.1f * pr.w);
      float gw = logf((m2 - m0) / pr.z) / 0.2f;
      float gh = logf((m3 - m1) / pr.w) / 0.2f;
      float4 ld = ((const float4*)loc)[(size_t)b * P_ + p];
      lossl = smooth_l1(ld.x - gx) + smooth_l1(ld.y - gy) +
              smooth_l1(ld.z - gw) + smooth_l1(ld.w - gh);
    }
  }

  redf[tid] = lossl; redf2[tid] = cep; redi[tid] = pc;
  __syncthreads();
  for (int s = TPB / 2; s > 0; s >>= 1) {
    if (tid < s) {
      redf[tid] += redf[tid + s];
      redf2[tid] += redf2[tid + s];
      redi[tid] += redi[tid + s];
    }
    __syncthreads();
  }
  if (tid == 0) {
    int bid = b * PCHUNKS + blockIdx.x;
    partial_l[bid] = redf[0];
    partial_c[bid] = redf2[0];
    atomicAdd(&num_pos[b], redi[0]);
  }
}

// ---------------------------------------------------------------------------
// Kernel 4: per-batch hard-negative mining. Exact top-k of loss_mine via
// 32-step radix search on float-as-uint keys held in LDS; ties share value T
// so their contribution is r * T (stable-argsort equivalent).
// ---------------------------------------------------------------------------
__global__ void k_negsel(const float* __restrict__ val,
                         const int* __restrict__ num_pos,
                         float* __restrict__ batch_c) {
  __shared__ unsigned int keys[P_];  // 34928 B
  __shared__ int ired[TPB];
  __shared__ float fred[TPB];
  const int b = blockIdx.x;
  const int tid = threadIdx.x;

  for (int i = tid; i < P_; i += TPB) {
    float v = val[(size_t)b * P_ + i];
    keys[i] = (v < 0.0f) ? 0u : __float_as_uint(v);  // positives -> loss_mine = 0
  }
  __syncthreads();

  int np = num_pos[b];
  int k = min(3 * np, P_ - 1);
  if (k <= 0) { if (tid == 0) batch_c[b] = 0.0f; return; }

  unsigned int prefix = 0u;
  for (int bit = 31; bit >= 0; --bit) {
    unsigned int cand = prefix | (1u << bit);
    int cnt = 0;
    for (int i = tid; i < P_; i += TPB) cnt += (keys[i] >= cand) ? 1 : 0;
    ired[tid] = cnt;
    __syncthreads();
    for (int s = TPB / 2; s > 0; s >>= 1) {
      if (tid < s) ired[tid] += ired[tid + s];
      __syncthreads();
    }
    int total = ired[0];
    __syncthreads();
    if (total >= k) prefix = cand;  // uniform across block
  }
  const unsigned int T = prefix;  // k-th largest key

  int cgt = 0; float sgt = 0.0f;
  for (int i = tid; i < P_; i += TPB) {
    unsigned int key = keys[i];
    if (key > T) { cgt++; sgt += __uint_as_float(key); }
  }
  ired[tid] = cgt; fred[tid] = sgt;
  __syncthreads();
  for (int s = TPB / 2; s > 0; s >>= 1) {
    if (tid < s) { ired[tid] += ired[tid + s]; fred[tid] += fred[tid + s]; }
    __syncthreads();
  }
  if (tid == 0) {
    int r = k - ired[0];  // tie slots, all of value T
    batch_c[b] = fred[0] + (float)r * __uint_as_float(T);
  }
}

// ---------------------------------------------------------------------------
// Kernel 5: single-wave finalize. WMMA f32 16x16x4 with all-ones B reduces
// 64 f32 partials per issue, accumulating in the C operand.
// A layout (ISA 7.12.2): lane l holds A[l%16][k0+r], k0 = (l<16?0:2).
// D layout: VGPR r -> row r (lanes 0-15) / row 8+r (lanes 16-31).
// ---------------------------------------------------------------------------
__device__ __forceinline__ float wmma_sum(const float* __restrict__ buf, int n) {
  const int lane = threadIdx.x & 31;
  const int row = lane & 15;
  const int k0 = (lane >> 4) << 1;
  v2f ones; ones.x = 1.0f; ones.y = 1.0f;
  v8f c = {};
  for (int base = 0; base < n; base += 64) {
    const float* p = buf + base + row * 4 + k0;
    v2f a; a.x = p[0]; a.y = p[1];
    c = __builtin_amdgcn_wmma_f32_16x16x4_f32(false, a, false, ones,
                                              (short)0, c, false, false);
  }
  float s = c.s0 + c.s1 + c.s2 + c.s3 + c.s4 + c.s5 + c.s6 + c.s7;
  return __shfl(s, 0, 32) + __shfl(s, 16, 32);  // rows 0-7 + rows 8-15
}

__global__ void k_final(const float* __restrict__ partial_l,
                        const float* __restrict__ partial_c,
                        const float* __restrict__ batch_c,
                        const int* __restrict__ num_pos,
                        float* __restrict__ out) {
  float sl = wmma_sum(partial_l, NBLK3_PAD);
  float sc = wmma_sum(partial_c, NBLK3_PAD);
  float sb = wmma_sum(batch_c, 128);
  int n = 0;
  for (int i = 0; i < B_; ++i) n += num_pos[i];
  float N = (float)n;
  if (threadIdx.x == 0) {
    out[0] = sl / N;
    out[1] = (sc + sb) / N;
  }
}

// ---------------------------------------------------------------------------
extern "C" void kernel_launch(void* const* d_in, const int* in_sizes, int n_in,
                              void* d_out, int out_size, void* d_ws, size_t ws_size,
                              hipStream_t stream) {
  const float* loc = (const float*)d_in[0];
  const float* conf = (const float*)d_in[1];
  const float* priors = (const float*)d_in[2];
  const float* targets = (const float*)d_in[3];
  float* out = (float*)d_out;

  char* ws = (char*)d_ws;
  size_t off = 0;
  float* partial_l = (float*)(ws + off); off += (size_t)NBLK3_PAD * 4;
  float* partial_c = (float*)(ws + off); off += (size_t)NBLK3_PAD * 4;
  float* batch_c = (float*)(ws + off);   off += 128 * 4;
  int* num_pos = (int*)(ws + off);       off += 128 * 4;
  unsigned long long* bpi = (unsigned long long*)(ws + off); off += (size_t)B_ * M_ * 8;
  const size_t accBytes = off;  // zeroed every call (capture-safe)
  float* bto = (float*)(ws + off); off += (size_t)B_ * P_ * 4;
  int* bti = (int*)(ws + off);     off += (size_t)B_ * P_ * 4;
  float* val = (float*)(ws + off); off += (size_t)B_ * P_ * 4;

  (void)hipMemsetAsync(d_ws, 0, accBytes, stream);

  dim3 grid(PCHUNKS, B_);
  k_match<<<grid, TPB, 0, stream>>>(priors, targets, bto, bti, bpi);
  k_force<<<1, 128, 0, stream>>>(bpi, bto, bti);
  k_main<<<grid, TPB, 0, stream>>>(loc, conf, priors, targets, bto, bti, val,
                                   partial_l, partial_c, num_pos);
  k_negsel<<<B_, TPB, 0, stream>>>(val, num_pos, batch_c);
  k_final<<<1, 32, 0, stream>>>(partial_l, partial_c, batch_c, num_pos, out);
}